// PiecewiseBlur_82463372083724
// MI455X (gfx1250) — compile-verified
//
#include <hip/hip_runtime.h>

// CDNA5 / gfx1250, wave32. Depthwise 15x15 per-tile blur via V_WMMA_F32_16X16X4_F32.
//
// Math note: reference's cosine-window blend cancels exactly (non-overlapping
// tiles, out = (conv*w)/w), so we emit the conv result directly.
//
// WMMA formulation: A (16x16, zero-padded kernel, row=dy col=dx) x
// B (Hankel window of one input row) -> D[dy,n] = horizontal conv of that row
// against all 15 kernel rows at once. Each D[dy,n] belongs to exactly one
// output row y = r - dy, scattered branch-free into a trash-padded LDS
// accumulator with unconditional ds_add_f32 (dy=15 row of A is zero, and
// out-of-band rows land in trash rows and are discarded).

typedef __attribute__((ext_vector_type(2))) float v2f;
typedef __attribute__((ext_vector_type(8))) float v8f;

#define TILE   256
#define BY     64                  // output rows per block
#define BX     64                  // output cols per block
#define IPH    (BY + 14)           // 78 halo-included input rows
#define IPW    (BX + 16)           // 80 halo-included input cols
#define NCHAIN (IPH * (BX / 16))   // 312 WMMA chains per block (312 % 8 waves == 0)
#define OPAD   15                  // trash rows above accumulator (yl down to -15)
#define OROWS  (OPAD + BY + 14)    // 93 rows: absorbs yl in [-15, 77]

// ---------------------------------------------------------------------------
// Normalize the 16 (4x4) 15x15 kernels and store each as a zero-padded 16x16
// matrix (row=dy, col=dx) ready to be the WMMA A-matrix.
// ---------------------------------------------------------------------------
__global__ __launch_bounds__(256) void norm_kernels_k(const float* __restrict__ k,
                                                      float* __restrict__ knorm) {
    const int t   = blockIdx.x;    // tile id 0..15  (j*4 + i)
    const int tid = threadIdx.x;   // 0..255

    __shared__ float red[256];
    float v = 0.0f;
    if (tid < 225) v = k[t * 225 + tid];
    red[tid] = v;
    __syncthreads();
    for (int s = 128; s > 0; s >>= 1) {
        if (tid < s) red[tid] += red[tid + s];
        __syncthreads();
    }
    const float inv = 1.0f / (red[0] + 1e-12f);

    const int dy = tid >> 4;
    const int dx = tid & 15;
    const float val = (dy < 15 && dx < 15) ? k[t * 225 + dy * 15 + dx] * inv : 0.0f;
    knorm[t * 256 + tid] = val;   // padded row/col 15 are zero
}

// ---------------------------------------------------------------------------
// Main kernel: one block = one 64x64 output sub-tile of one (batch,ch,tile).
// ---------------------------------------------------------------------------
__global__ __launch_bounds__(256) void pwblur_k(const float* __restrict__ x,
                                                const float* __restrict__ knorm,
                                                float* __restrict__ out) {
    __shared__ float ip[IPH * IPW];      // 24,960 B: replicate-padded input band
    __shared__ float outl[OROWS * BX];   // 23,808 B: trash-padded accumulator

    const int tid  = threadIdx.x;
    const int lane = tid & 31;
    const int wave = tid >> 5;           // 0..7
    const bool hi  = (lane & 16) != 0;
    const int  nl  = lane & 15;          // N index within 16-wide x block

    // ---- decode block id -> (batch, ch, tile j, tile i, sub-band y, x) ----
    int bid   = blockIdx.x;              // 0..6143
    int sub   = bid & 15;                // 16 sub-blocks per tile
    int t     = bid >> 4;                // 0..383
    int ti    = t & 3;
    int tj    = (t >> 2) & 3;
    int plane = t >> 4;                  // 0..23 = batch*3 + ch
    int sub_y = sub >> 2;
    int sub_x = sub & 3;

    const int ty0 = tj * TILE;           // tile origin in image
    const int tx0 = ti * TILE;
    const int y0  = sub_y * BY;          // sub-block origin within tile
    const int x0  = sub_x * BX;

    const float* __restrict__ xin = x + (size_t)plane * 1024 * 1024;

    // ---- stage replicate-padded input band into LDS ----
    for (int idx = tid; idx < IPH * IPW; idx += 256) {
        int r  = idx / IPW;
        int xi = idx - r * IPW;
        int ys = y0 + r - 7;  ys = ys < 0 ? 0 : (ys > TILE - 1 ? TILE - 1 : ys);
        int xs = x0 + xi - 7; xs = xs < 0 ? 0 : (xs > TILE - 1 ? TILE - 1 : xs);
        ip[idx] = xin[(size_t)(ty0 + ys) * 1024 + (tx0 + xs)];
    }
    for (int idx = tid; idx < OROWS * BX; idx += 256) outl[idx] = 0.0f;

    // ---- load the WMMA A-matrix (this tile's padded kernel) into regs ----
    // A 16x4 layout (ISA 7.12.2): lanes 0-15 -> M=lane, VGPR0=K0 VGPR1=K1;
    // lanes 16-31 -> M=lane-16, VGPR0=K2 VGPR1=K3.  Chunk c covers dx=4c..4c+3.
    const float* __restrict__ kn = knorm + (tj * 4 + ti) * 256;
    v2f a[4];
    {
        const int m = nl;
        #pragma unroll
        for (int c = 0; c < 4; ++c) {
            int dxb = 4 * c + (hi ? 2 : 0);
            a[c].x = kn[m * 16 + dxb];
            a[c].y = kn[m * 16 + dxb + 1];
        }
    }
    __syncthreads();

    // ---- WMMA chains: 312 chains, 39 per wave, uniform (EXEC all ones) ----
    for (int cid = wave; cid < NCHAIN; cid += 8) {
        const int r  = cid >> 2;           // input row index in band, 0..77
        const int xb = (cid & 3) * 16;     // x block origin, 0/16/32/48

        // B 4x16 layout (mirror of A): VGPR0 lanes0-15 = B[K=0,n], lanes16-31
        // = B[K=2,n]; VGPR1 = K1 / K3.  B[k,n] = ip[r][xb + n + 4c + k].
        const int bbase = r * IPW + xb + nl + (hi ? 2 : 0);

        v8f acc = {};
        #pragma unroll
        for (int c = 0; c < 4; ++c) {
            v2f bv;
            bv.x = ip[bbase + 4 * c];
            bv.y = ip[bbase + 4 * c + 1];
            acc = __builtin_amdgcn_wmma_f32_16x16x4_f32(
                false, a[c], false, bv, (short)0, acc, false, false);
        }

        // D 16x16 f32 layout: VGPR i -> M=i (lanes 0-15) / M=8+i (lanes 16-31).
        // Row dy = i + 8*hi of D belongs to output row yl = r - dy.
        // Branch-free scatter: yl+OPAD always lands inside the padded
        // accumulator (trash rows absorb out-of-band; dy=15 adds exact 0.0).
        const int sbase = (r + OPAD - (hi ? 8 : 0)) * BX + xb + nl;
        #pragma unroll
        for (int i = 0; i < 8; ++i) {
            atomicAdd(&outl[sbase - i * BX], acc[i]);
        }
    }
    __syncthreads();

    // ---- write back (skip trash rows) ----
    float* __restrict__ op = out + (size_t)plane * 1024 * 1024;
    for (int idx = tid; idx < BY * BX; idx += 256) {
        int yl = idx >> 6;          // /BX
        int xl = idx & 63;          // %BX
        op[(size_t)(ty0 + y0 + yl) * 1024 + (tx0 + x0 + xl)] = outl[(OPAD + yl) * BX + xl];
    }
}

// ---------------------------------------------------------------------------
extern "C" void kernel_launch(void* const* d_in, const int* in_sizes, int n_in,
                              void* d_out, int out_size, void* d_ws, size_t ws_size,
                              hipStream_t stream) {
    const float* x    = (const float*)d_in[0];   // (8,3,1024,1024) f32
    const float* kern = (const float*)d_in[1];   // (4,4,15,15) f32
    float* outp  = (float*)d_out;                // (8,3,1024,1024) f32
    float* knorm = (float*)d_ws;                 // 16 tiles * 256 floats = 16 KB

    norm_kernels_k<<<16, 256, 0, stream>>>(kern, knorm);

    // 8 batch * 3 ch * 16 tiles * 16 sub-blocks = 6144 blocks, 256 thr (8 waves)
    pwblur_k<<<6144, 256, 0, stream>>>(x, knorm, outp);
}